// GCNNet2_42812234006621
// MI455X (gfx1250) — compile-verified
//
#include <hip/hip_runtime.h>
#include <hip/hip_bf16.h>
#include <math.h>

typedef __attribute__((ext_vector_type(2))) float v2f;
typedef __attribute__((ext_vector_type(8))) float v8f;

#define NNODES 50000
#define NEDGES 800000
#define DIM    128
#define NLAYERS 4
#define EPSBN  1e-5f

// ---------------------------------------------------------------- utilities
__global__ void zero_kernel(float* __restrict__ p, int n) {
    int i = blockIdx.x * blockDim.x + threadIdx.x;
    if (i < n) p[i] = 0.0f;
}

// one thread per edge: accumulate out/in degrees as floats
__global__ void degree_kernel(const int* __restrict__ src, const int* __restrict__ dst,
                              float* __restrict__ deg_out, float* __restrict__ deg_in) {
    int e = blockIdx.x * blockDim.x + threadIdx.x;
    if (e < NEDGES) {
        atomicAdd(&deg_out[src[e]], 1.0f);
        atomicAdd(&deg_in [dst[e]], 1.0f);
    }
}

// in-place d[i] = rsqrt(max(d[i],1))
__global__ void invsqrt_kernel(float* __restrict__ d, int n) {
    int i = blockIdx.x * blockDim.x + threadIdx.x;
    if (i < n) d[i] = rsqrtf(fmaxf(d[i], 1.0f));
}

// ------------------------------------------------------- WMMA fp32 GEMM
// C[m][n] = ((rowscale?rs[m]:1) * A[m][:]) dot W[:][n] + bias[n], then *post[m]
// One wave computes one 16x16 tile; block = 8 waves = 16x128 output strip.
// Grid.x = NNODES/16 (= 3125 exactly, no tails; EXEC all-ones for WMMA).
__global__ void wmma_gemm_kernel(const float* __restrict__ A,
                                 const float* __restrict__ W,
                                 const float* __restrict__ bias,
                                 const float* __restrict__ rowscale,   // nullable
                                 const float* __restrict__ postscale,  // nullable
                                 float* __restrict__ Cout) {
    const int m0    = blockIdx.x << 4;
    const int wave  = threadIdx.x >> 5;       // 0..7 -> N tile
    const int lane  = threadIdx.x & 31;
    const int n0    = wave << 4;
    const int l16   = lane & 15;
    const int khalf = lane >> 4;              // 0: K+{0,1}  1: K+{2,3}

    const int   row  = m0 + l16;              // A-fragment row (M = lane%16)
    const float rs   = rowscale ? rowscale[row] : 1.0f;
    const float* __restrict__ arow = A + (size_t)row * DIM;

    v8f c = {};
    for (int k0 = 0; k0 < DIM; k0 += 4) {
        const int k = k0 + khalf * 2;
        v2f a, b;
        a.x = arow[k]     * rs;               // A 16x4 fragment (2 VGPRs)
        a.y = arow[k + 1] * rs;
        b.x = W[(size_t)(k    ) * DIM + n0 + l16];  // B 4x16 fragment
        b.y = W[(size_t)(k + 1) * DIM + n0 + l16];
        c = __builtin_amdgcn_wmma_f32_16x16x4_f32(
                /*neg_a=*/false, a, /*neg_b=*/false, b,
                /*c_mod=*/(short)0, c, /*reuse_a=*/false, /*reuse_b=*/false);
    }

    const float bn = bias[n0 + l16];
#pragma unroll
    for (int r = 0; r < 8; ++r) {
        const int rowOut = m0 + r + khalf * 8;     // C/D layout: VGPR r -> M=r (+8)
        float v = c[r] + bn;
        if (postscale) v *= postscale[rowOut];
        Cout[(size_t)rowOut * DIM + n0 + l16] = v;
    }
}

// -------------------------------------------------- edge scatter (message agg)
// one wave per edge, each lane moves 4 consecutive floats (float4 = 16B coalesced)
__global__ void scatter_kernel(const int* __restrict__ src, const int* __restrict__ dst,
                               const float* __restrict__ h, const float* __restrict__ inv_out,
                               float* __restrict__ agg) {
    const int t    = blockIdx.x * blockDim.x + threadIdx.x;
    const int e    = t >> 5;
    const int lane = t & 31;
    if (e >= NEDGES) return;
    const int   s = src[e];
    const int   d = dst[e];
    const float w = inv_out[s];
    const float4 v = *(const float4*)(h + (size_t)s * DIM + lane * 4);
    float* __restrict__ out = agg + (size_t)d * DIM + lane * 4;
    atomicAdd(out + 0, v.x * w);
    atomicAdd(out + 1, v.y * w);
    atomicAdd(out + 2, v.z * w);
    atomicAdd(out + 3, v.w * w);
}

// -------------------------------------------------- BatchNorm column statistics
// block handles a 128-row chunk; thread owns (column = t&127, row-parity = t>>7)
__global__ void colstats_kernel(const float* __restrict__ t,
                                float* __restrict__ sum, float* __restrict__ sumsq) {
    const int c    = threadIdx.x & 127;
    const int half = threadIdx.x >> 7;
    const int rowBase = blockIdx.x * 128;
    float s = 0.0f, q = 0.0f;
    for (int r = half; r < 128; r += 2) {
        const int row = rowBase + r;
        if (row < NNODES) {
            const float v = t[(size_t)row * DIM + c];
            s += v;
            q += v * v;
        }
    }
    atomicAdd(&sum[c],   s);
    atomicAdd(&sumsq[c], q);
}

__global__ void bn_finalize_kernel(const float* __restrict__ sum, const float* __restrict__ sumsq,
                                   const float* __restrict__ gamma, const float* __restrict__ beta,
                                   float* __restrict__ scale, float* __restrict__ shift) {
    const int c = threadIdx.x;            // 128 threads
    const float inv_n = 1.0f / (float)NNODES;
    const float mu  = sum[c] * inv_n;
    const float var = sumsq[c] * inv_n - mu * mu;   // population variance (ddof=0)
    const float rsg = rsqrtf(var + EPSBN);
    const float sc  = rsg * gamma[c];
    scale[c] = sc;
    shift[c] = beta[c] - mu * sc;
}

// h[i] = h[i] + relu(t[i]*scale[col] + shift[col]) — in-place residual
__global__ void bn_relu_res_kernel(const float* __restrict__ t,
                                   const float* __restrict__ scale,
                                   const float* __restrict__ shift,
                                   float* __restrict__ h) {
    const int i    = blockIdx.x * blockDim.x + threadIdx.x;   // one float4
    const int base = i * 4;
    if (base >= NNODES * DIM) return;
    const int c = base & 127;
    const float4 tv = *(const float4*)(t + base);
    float4 hv = *(const float4*)(h + base);
    hv.x += fmaxf(tv.x * scale[c + 0] + shift[c + 0], 0.0f);
    hv.y += fmaxf(tv.y * scale[c + 1] + shift[c + 1], 0.0f);
    hv.z += fmaxf(tv.z * scale[c + 2] + shift[c + 2], 0.0f);
    hv.w += fmaxf(tv.w * scale[c + 3] + shift[c + 3], 0.0f);
    *(float4*)(h + base) = hv;
}

// ---------------------------------------------------------------- readout
// wave per row: 32 lanes x float4 = 128-wide dot; wave32 shuffle reduction
__global__ void readout_kernel(const float* __restrict__ h, const float* __restrict__ w,
                               float* __restrict__ out) {
    const int t    = blockIdx.x * blockDim.x + threadIdx.x;
    const int row  = t >> 5;
    const int lane = t & 31;
    if (row >= NNODES) return;
    const float4 hv = *(const float4*)(h + (size_t)row * DIM + lane * 4);
    const float4 wv = *(const float4*)(w + lane * 4);
    float p = hv.x * wv.x + hv.y * wv.y + hv.z * wv.z + hv.w * wv.w;
#pragma unroll
    for (int o = 16; o > 0; o >>= 1) p += __shfl_down(p, o, 32);
    if (lane == 0) {
        out[row]          = p;
        out[NNODES + row] = 1.0f / (1.0f + expf(-p));
    }
}

// ---------------------------------------------------------------- driver
extern "C" void kernel_launch(void* const* d_in, const int* in_sizes, int n_in,
                              void* d_out, int out_size, void* d_ws, size_t ws_size,
                              hipStream_t stream) {
    const float* nodes_feat = (const float*)d_in[0];
    const float* snorm      = (const float*)d_in[1];   // nodes_num_norm_sqrt, [N]
    const int*   src        = (const int*)  d_in[2];
    const int*   dst        = (const int*)  d_in[3];
    const float* emb_w      = (const float*)d_in[4];
    const float* emb_b      = (const float*)d_in[5];
    const float* Ws         = (const float*)d_in[6];   // [L,D,D]
    const float* bs         = (const float*)d_in[7];   // [L,D]
    const float* gammas     = (const float*)d_in[8];
    const float* betas      = (const float*)d_in[9];
    const float* readout_w  = (const float*)d_in[10];
    float* out = (float*)d_out;                        // [2N]: logits | sigmoid

    // workspace layout
    float* ws       = (float*)d_ws;
    float* h        = ws;                                  // N*D
    float* tmp      = h   + (size_t)NNODES * DIM;          // N*D
    float* agg      = tmp + (size_t)NNODES * DIM;          // N*D
    float* inv_out  = agg + (size_t)NNODES * DIM;          // N (deg_out -> rsqrt)
    float* inv_in   = inv_out + NNODES;                    // N (deg_in  -> rsqrt)
    float* colsum   = inv_in  + NNODES;                    // D
    float* colsumsq = colsum  + DIM;                       // D
    float* scale    = colsumsq + DIM;                      // D
    float* shift    = scale    + DIM;                      // D

    const int B = 256;

    // degrees -> inverse sqrt (inv_out/inv_in are contiguous: one pass each)
    zero_kernel   <<<(2 * NNODES + B - 1) / B, B, 0, stream>>>(inv_out, 2 * NNODES);
    degree_kernel <<<(NEDGES + B - 1) / B,     B, 0, stream>>>(src, dst, inv_out, inv_in);
    invsqrt_kernel<<<(2 * NNODES + B - 1) / B, B, 0, stream>>>(inv_out, 2 * NNODES);

    // embedding: h = X @ emb_w + emb_b
    wmma_gemm_kernel<<<NNODES / 16, B, 0, stream>>>(nodes_feat, emb_w, emb_b,
                                                    nullptr, nullptr, h);

    for (int l = 0; l < NLAYERS; ++l) {
        // agg = segment_sum over edges of (h * inv_sqrt_out)[src]
        zero_kernel   <<<(NNODES * DIM + B - 1) / B, B, 0, stream>>>(agg, NNODES * DIM);
        scatter_kernel<<<(NEDGES * 32) / B,          B, 0, stream>>>(src, dst, h, inv_out, agg);

        // tmp = ((agg * inv_sqrt_in) @ W_l + b_l) * snorm
        wmma_gemm_kernel<<<NNODES / 16, B, 0, stream>>>(agg,
                                                        Ws + (size_t)l * DIM * DIM,
                                                        bs + (size_t)l * DIM,
                                                        inv_in, snorm, tmp);

        // BatchNorm batch statistics over all N rows
        zero_kernel       <<<1, 2 * DIM, 0, stream>>>(colsum, 2 * DIM);
        colstats_kernel   <<<(NNODES + 127) / 128, B, 0, stream>>>(tmp, colsum, colsumsq);
        bn_finalize_kernel<<<1, DIM, 0, stream>>>(colsum, colsumsq,
                                                  gammas + (size_t)l * DIM,
                                                  betas  + (size_t)l * DIM,
                                                  scale, shift);

        // h = h + relu(BN(tmp))  (in-place residual)
        bn_relu_res_kernel<<<(NNODES * DIM / 4) / B, B, 0, stream>>>(tmp, scale, shift, h);
    }

    // logits + sigmoid
    readout_kernel<<<(NNODES * 32) / B, B, 0, stream>>>(h, readout_w, out);
}